// ConvCapsuleLayer_4758823764535
// MI455X (gfx1250) — compile-verified
//
#include <hip/hip_runtime.h>

typedef __attribute__((ext_vector_type(2))) float v2f;
typedef __attribute__((ext_vector_type(8))) float v8f;

#define BATCH    2
#define HIN      48
#define WIN      48
#define IN_CAPS  8
#define ATOMS    16
#define KER      3
#define OUT_CAPS 16
#define ROUTINGS 3
#define HO       46
#define WO       46
#define NPATCH   (BATCH * HO * WO)        // 4232
#define KKIN     (KER * KER * IN_CAPS)    // 72
#define CIN      (IN_CAPS * (ATOMS + 1))  // 136

#define PPW 4                 // patches per wave
#define WPB 4                 // waves per block
#define PPB (PPW * WPB)       // 16 patches per block

// per-wave LDS layout (floats)
#define POSE_F (PPW * KKIN * 16)   // 4608
#define ACT_F  (PPW * KKIN)        // 288
#define LOG_F  (PPW * KKIN * 16)   // 4608
#define COUP_F (PPW * KKIN * 16)   // 4608
#define WAVE_F (POSE_F + ACT_F + LOG_F + COUP_F)   // 14112
#define WLDS_F (IN_CAPS * OUT_CAPS * 16)           // 2048
#define SMEM_F (WLDS_F + WPB * WAVE_F)
#define SMEM_BYTES (SMEM_F * 4)                    // 233984 B

__global__ void __launch_bounds__(WPB * 32)
caps_route_kernel(const float* __restrict__ x,
                  const float* __restrict__ Wg,
                  float* __restrict__ out)
{
    extern __shared__ float smem[];
    float* Wl = smem;

    const int tid  = threadIdx.x;
    const int lane = tid & 31;
    const int wave = tid >> 5;

    float* pose   = smem + WLDS_F + wave * WAVE_F;
    float* act    = pose + POSE_F;
    float* logits = act + ACT_F;
    float* coup   = logits + LOG_F;

    // ---- cooperative load of W (raw layout [i][o][b][c]) ----
    for (int i = tid; i < WLDS_F; i += WPB * 32) Wl[i] = Wg[i];

    const int patch_base = blockIdx.x * PPB + wave * PPW;

    // ---- load pose rows + activations into LDS (clamped for tail) ----
    for (int r = lane; r < PPW * KKIN; r += 32) {
        const int p  = r / KKIN;
        const int ki = r % KKIN;
        const int kk = ki >> 3;     // kernel position
        const int ic = ki & 7;      // input capsule
        const int di = kk / 3, dj = kk % 3;
        int n = patch_base + p; if (n >= NPATCH) n = NPATCH - 1;
        const int wo = n % WO;
        const int t1 = n / WO;
        const int ho = t1 % HO;
        const int bb = t1 / HO;
        const float* src = x + ((size_t)((bb * HIN + ho + di) * WIN) + (wo + dj)) * CIN
                             + ic * (ATOMS + 1);
        float* dst = pose + r * 16;
        #pragma unroll
        for (int e = 0; e < 16; ++e) dst[e] = src[e];
        act[r] = src[ATOMS];
    }
    for (int i = lane; i < LOG_F; i += 32) logits[i] = 0.0f;
    __syncthreads();

    // ---- per-lane WMMA index decode ----
    // C/D: lanes 0-15 -> M = vgpr j ; lanes 16-31 -> M = j+8 ; N = lane%16
    // A (16x4 f32): lanes 0-15 K=0,1 ; lanes 16-31 K=2,3 ; M = lane%16
    // B (4x16 f32): lanes 0-15 K=0,1 ; lanes 16-31 K=2,3 ; N = lane%16
    const int half   = lane >> 4;
    const int nloc   = lane & 15;
    const int olocal = nloc >> 2;   // out-cap within block of 4
    const int cc     = nloc & 3;    // vote column c
    const int p0 = half * 2, p1 = p0 + 1;   // the two patches this lane's D covers
    const int pa = nloc >> 2;       // A-row patch
    const int aa = nloc & 3;        // A-row 'a'
    const int b0 = half * 2;        // K pair (pose col b)

    for (int r = 0; r < ROUTINGS; ++r) {
        // ---- coup = softmax(logits over o) * act ----
        for (int row = lane; row < PPW * KKIN; row += 32) {
            const float* lg = logits + row * 16;
            float m = lg[0];
            #pragma unroll
            for (int o = 1; o < 16; ++o) m = fmaxf(m, lg[o]);
            float e[16], sum = 0.0f;
            #pragma unroll
            for (int o = 0; o < 16; ++o) { e[o] = __expf(lg[o] - m); sum += e[o]; }
            const float sc = act[row] / sum;
            float* cp = coup + row * 16;
            #pragma unroll
            for (int o = 0; o < 16; ++o) cp[o] = e[o] * sc;
        }
        __syncthreads();

        // ---- s-pass: recompute votes via WMMA, accumulate coup-weighted ----
        v8f vreg[4];
        for (int ob = 0; ob < 4; ++ob) {
            const int o = ob * 4 + olocal;
            v8f s = {0, 0, 0, 0, 0, 0, 0, 0};
            for (int ic = 0; ic < IN_CAPS; ++ic) {
                const int wbase = ((ic * OUT_CAPS + o) * 4) * 4 + cc; // [i][o][b][c]
                v2f Bv;
                Bv.x = Wl[wbase + b0 * 4];
                Bv.y = Wl[wbase + (b0 + 1) * 4];
                for (int kk = 0; kk < KER * KER; ++kk) {
                    const int ki = kk * IN_CAPS + ic;
                    const float* ap = pose + ((pa * KKIN + ki) * 16 + aa * 4 + b0);
                    v2f Av; Av.x = ap[0]; Av.y = ap[1];
                    v8f zero = {0, 0, 0, 0, 0, 0, 0, 0};
                    v8f D = __builtin_amdgcn_wmma_f32_16x16x4_f32(
                        false, Av, false, Bv, (short)0, zero, false, false);
                    const float c0 = coup[(p0 * KKIN + ki) * 16 + o];
                    const float c1 = coup[(p1 * KKIN + ki) * 16 + o];
                    #pragma unroll
                    for (int j = 0; j < 4; ++j) s[j] += c0 * D[j];
                    #pragma unroll
                    for (int j = 4; j < 8; ++j) s[j] += c1 * D[j];
                }
            }
            // ---- squash: norm over e=(a,c); a in-lane, c via 4-lane shfl ----
            float sq0 = s[0]*s[0] + s[1]*s[1] + s[2]*s[2] + s[3]*s[3];
            float sq1 = s[4]*s[4] + s[5]*s[5] + s[6]*s[6] + s[7]*s[7];
            sq0 += __shfl_xor(sq0, 1); sq0 += __shfl_xor(sq0, 2);
            sq1 += __shfl_xor(sq1, 1); sq1 += __shfl_xor(sq1, 2);
            const float k0 = (sq0 / (1.0f + sq0)) * rsqrtf(sq0 + 1e-7f);
            const float k1 = (sq1 / (1.0f + sq1)) * rsqrtf(sq1 + 1e-7f);
            v8f v;
            #pragma unroll
            for (int j = 0; j < 4; ++j) v[j] = s[j] * k0;
            #pragma unroll
            for (int j = 4; j < 8; ++j) v[j] = s[j] * k1;
            vreg[ob] = v;
        }

        if (r == ROUTINGS - 1) {
            // ---- final: write v straight from registers ----
            const int n0 = patch_base + p0;
            const int n1 = patch_base + p1;
            #pragma unroll
            for (int ob = 0; ob < 4; ++ob) {
                const int o = ob * 4 + olocal;
                if (n0 < NPATCH) {
                    #pragma unroll
                    for (int j = 0; j < 4; ++j)
                        out[(size_t)n0 * 256 + o * 16 + j * 4 + cc] = vreg[ob][j];
                }
                if (n1 < NPATCH) {
                    #pragma unroll
                    for (int j = 0; j < 4; ++j)
                        out[(size_t)n1 * 256 + o * 16 + j * 4 + cc] = vreg[ob][4 + j];
                }
            }
        } else {
            // ---- logits += <votes, v> : recompute votes via WMMA ----
            for (int ob = 0; ob < 4; ++ob) {
                const int o = ob * 4 + olocal;
                const v8f v = vreg[ob];
                for (int ic = 0; ic < IN_CAPS; ++ic) {
                    const int wbase = ((ic * OUT_CAPS + o) * 4) * 4 + cc;
                    v2f Bv;
                    Bv.x = Wl[wbase + b0 * 4];
                    Bv.y = Wl[wbase + (b0 + 1) * 4];
                    for (int kk = 0; kk < KER * KER; ++kk) {
                        const int ki = kk * IN_CAPS + ic;
                        const float* ap = pose + ((pa * KKIN + ki) * 16 + aa * 4 + b0);
                        v2f Av; Av.x = ap[0]; Av.y = ap[1];
                        v8f zero = {0, 0, 0, 0, 0, 0, 0, 0};
                        v8f D = __builtin_amdgcn_wmma_f32_16x16x4_f32(
                            false, Av, false, Bv, (short)0, zero, false, false);
                        float t0 = D[0]*v[0] + D[1]*v[1] + D[2]*v[2] + D[3]*v[3];
                        float t1 = D[4]*v[4] + D[5]*v[5] + D[6]*v[6] + D[7]*v[7];
                        t0 += __shfl_xor(t0, 1); t0 += __shfl_xor(t0, 2);
                        t1 += __shfl_xor(t1, 1); t1 += __shfl_xor(t1, 2);
                        if (cc == 0) {   // unique writer per (p,ki,o)
                            logits[(p0 * KKIN + ki) * 16 + o] += t0;
                            logits[(p1 * KKIN + ki) * 16 + o] += t1;
                        }
                    }
                }
            }
            __syncthreads();
        }
    }
}

extern "C" void kernel_launch(void* const* d_in, const int* in_sizes, int n_in,
                              void* d_out, int out_size, void* d_ws, size_t ws_size,
                              hipStream_t stream) {
    (void)in_sizes; (void)n_in; (void)d_ws; (void)ws_size; (void)out_size;
    const float* x = (const float*)d_in[0];
    const float* W = (const float*)d_in[1];
    float* out = (float*)d_out;
    dim3 grid((NPATCH + PPB - 1) / PPB);   // 265 blocks
    dim3 block(WPB * 32);                  // 128 threads = 4 waves (wave32)
    caps_route_kernel<<<grid, block, SMEM_BYTES, stream>>>(x, W, out);
}